// KuramotoCell_14714557956639
// MI455X (gfx1250) — compile-verified
//
#include <hip/hip_runtime.h>
#include <math.h>

// Kuramoto cell, fused single kernel for gfx1250 (MI455X).
//
// coupling[b,i] = sum_j Wh[i,j] * sin(state[b,i] - state[b,j])
//              = sin(s_bi) * P[b,i] - cos(s_bi) * Q[b,i]
//   where P = cos(state) @ Wh^T,  Q = sin(state) @ Wh^T   (two GEMMs, shared B operand)
// out = mod(coupling + x@Wi_w^T + Wi_b + omega + state, 2*pi)
//
// Wh (16 MB, the dominant HBM stream) is staged into LDS by the Tensor Data
// Mover (tensor_load_to_lds, TENSORcnt), double-buffered against the WMMA
// compute phase. sin/cos(state) tiles are produced by VALU and ds_store'd.

typedef float v2f __attribute__((ext_vector_type(2)));
typedef float v4f __attribute__((ext_vector_type(4)));
typedef float v8f __attribute__((ext_vector_type(8)));
typedef unsigned int u32x4 __attribute__((ext_vector_type(4)));
typedef int i32x4 __attribute__((ext_vector_type(4)));
typedef int i32x8 __attribute__((ext_vector_type(8)));

#define TWO_PI_F     6.283185307179586f
#define INV_TWO_PI_F 0.15915494309189535f

// Tiling: workgroup = 256 threads = 8 waves (wave32).
//   N-block per workgroup : 64 hidden columns (4 n-tiles of 16)
//   M                     : 32 batch rows (2 m-tiles of 16)
//   wave w -> (m_tile = w & 1, n_tile = w >> 1)
//   K-block               : 32 (8 WMMA k-steps of 4), double-buffered
#define BK   32
#define BKP  36   // padded LDS row stride (floats); TDM pad: every 32 dw insert 4 dw
#define NBLK 64

// Issue a TDM load of a [64 rows x 32 dwords] Wh tile (row stride n_hid floats)
// into LDS at lds_byte_off, with hardware padding to a 36-dword row stride.
// D# layout per CDNA5 ISA section 8.3/8.4.
__device__ __forceinline__ void tdm_load_wh_tile(const float* gaddr,
                                                 unsigned lds_byte_off,
                                                 int n_hid) {
    unsigned long long ga = (unsigned long long)(uintptr_t)gaddr;

    u32x4 g0;
    g0[0] = 1u;                                   // count=1 (valid), user mode, no gather
    g0[1] = lds_byte_off;                         // lds_addr (bytes)
    g0[2] = (unsigned)(ga & 0xffffffffu);         // global_addr[31:0]
    g0[3] = (unsigned)((ga >> 32) & 0x01ffffffu)  // global_addr[56:32]
            | (2u << 30);                         // type = 2 ("image")

    const unsigned dim = (unsigned)n_hid;         // tensor dims (OOB bound only)
    i32x8 g1;
    g1[0] = (int)((2u << 16)                      // data_size   = 2 -> 4 bytes
                | (1u << 20)                      // pad_enable  = 1
                | (4u << 22)                      // pad_interval= 4 -> every 32 dwords
                | (3u << 25));                    // pad_amount  = 3 -> insert 4 dwords
    g1[1] = (int)(dim << 16);                     // tensor_dim0[15:0]   (bits 79:48)
    g1[2] = (int)((dim >> 16) | (dim << 16));     // tensor_dim0[31:16] | tensor_dim1[15:0]
    g1[3] = (int)((dim >> 16) | (32u << 16));     // tensor_dim1[31:16] | tile_dim0 = 32
    g1[4] = 64;                                   // tile_dim1 = 64 rows, tile_dim2 = 0
    g1[5] = n_hid;                                // tensor_dim0_stride[31:0] (dword units)
    g1[6] = 0;                                    // stride hi | tensor_dim1_stride lo
    g1[7] = 0;

    i32x4 z4 = {0, 0, 0, 0};                      // groups 2/3 unused for a 2-D tile
    i32x8 z8 = {0, 0, 0, 0, 0, 0, 0, 0};          // extra group (clang-23 6-arg form)
    __builtin_amdgcn_tensor_load_to_lds(g0, g1, z4, z4, z8, 0);
}

__global__ __launch_bounds__(256, 1)
void kuramoto_wmma_kernel(const float* __restrict__ x,
                          const float* __restrict__ state,
                          const float* __restrict__ Wi_w,
                          const float* __restrict__ Wi_b,
                          const float* __restrict__ Wh,
                          const float* __restrict__ omega,
                          float* __restrict__ out,
                          int Bsz, int n_inp, int n_hid) {
    __shared__ float Ac[2][32 * BKP];    // cos(state) tiles [32 x BK], double-buffered
    __shared__ float As[2][32 * BKP];    // sin(state) tiles
    __shared__ float Ws[2][NBLK * BKP];  // Wh tiles [64 x BK] via TDM, double-buffered

    const int tid  = threadIdx.x;
    const int lane = tid & 31;
    const int wave = tid >> 5;
    const int mt   = wave & 1;           // m-tile: batch rows [mt*16, mt*16+16)
    const int nt   = wave >> 1;          // n-tile within the 64-wide N block
    const int n0   = blockIdx.x * NBLK;

    v8f accP = {0.f, 0.f, 0.f, 0.f, 0.f, 0.f, 0.f, 0.f};  // cosA @ Wh^T
    v8f accQ = {0.f, 0.f, 0.f, 0.f, 0.f, 0.f, 0.f, 0.f};  // sinA @ Wh^T

    const int nkb = n_hid / BK;          // 64 K-blocks for n_hid = 2048

    // state-tile staging map (256 threads): 32 rows x 8 float4-chunks
    const int ar  = tid >> 3;            // 0..31 (batch row)
    const int ac4 = (tid & 7) << 2;      // 0..28 (k offset, float4)

    const float* whb = Wh + (size_t)n0 * n_hid;

    // ---- prologue: stage K-block 0 into buffer 0
    if (wave == 0) {
        tdm_load_wh_tile(whb, (unsigned)(uintptr_t)&Ws[0][0], n_hid);
    }
    {
        v4f s = *(const v4f*)(state + ar * n_hid + ac4);
        float sn0, cs0, sn1, cs1, sn2, cs2, sn3, cs3;
        __sincosf(s.x, &sn0, &cs0);
        __sincosf(s.y, &sn1, &cs1);
        __sincosf(s.z, &sn2, &cs2);
        __sincosf(s.w, &sn3, &cs3);
        v4f cv = {cs0, cs1, cs2, cs3};
        v4f sv = {sn0, sn1, sn2, sn3};
        *(v4f*)(&Ac[0][ar * BKP + ac4]) = cv;
        *(v4f*)(&As[0][ar * BKP + ac4]) = sv;
    }
    if (wave == 0) {
        __builtin_amdgcn_s_wait_tensorcnt(0);
    }
    __syncthreads();

    // WMMA fragment addressing (f32 16x16x4):
    //   A: lane 0-15 holds M=lane, K={4k,4k+1}; lane 16-31 holds K={4k+2,4k+3}
    //   B: lane 0-15 holds N=lane, K={4k,4k+1}; lane 16-31 holds K={4k+2,4k+3}
    const int frow_a = mt * 16 + (lane & 15);
    const int frow_b = nt * 16 + (lane & 15);
    const int khalf  = (lane >> 4) << 1;  // 0 or 2

    for (int kb = 0; kb < nkb; ++kb) {
        const int cur = kb & 1;
        const int nxt = cur ^ 1;

        // ---- stage K-block kb+1 into the other buffer (overlapped with WMMA)
        if (kb + 1 < nkb) {
            const int k1 = (kb + 1) * BK;
            if (wave == 0) {
                tdm_load_wh_tile(whb + k1, (unsigned)(uintptr_t)&Ws[nxt][0], n_hid);
            }
            v4f s = *(const v4f*)(state + ar * n_hid + k1 + ac4);
            float sn0, cs0, sn1, cs1, sn2, cs2, sn3, cs3;
            __sincosf(s.x, &sn0, &cs0);
            __sincosf(s.y, &sn1, &cs1);
            __sincosf(s.z, &sn2, &cs2);
            __sincosf(s.w, &sn3, &cs3);
            v4f cv = {cs0, cs1, cs2, cs3};
            v4f sv = {sn0, sn1, sn2, sn3};
            *(v4f*)(&Ac[nxt][ar * BKP + ac4]) = cv;
            *(v4f*)(&As[nxt][ar * BKP + ac4]) = sv;
            if (kb + 2 < nkb) {  // warm L2 for the K-block after next
                __builtin_prefetch(state + ar * n_hid + k1 + BK + ac4, 0, 1);
            }
        }

        // ---- 8 WMMA k-steps of 4 on the current buffers; P and Q share B
        const float* acb = &Ac[cur][frow_a * BKP];
        const float* asb = &As[cur][frow_a * BKP];
        const float* wsb = &Ws[cur][frow_b * BKP];
#pragma unroll
        for (int ks = 0; ks < BK / 4; ++ks) {
            const int kk = ks * 4 + khalf;
            v2f a_cos = *(const v2f*)(acb + kk);
            v2f a_sin = *(const v2f*)(asb + kk);
            v2f b_wh  = *(const v2f*)(wsb + kk);
            accP = __builtin_amdgcn_wmma_f32_16x16x4_f32(
                false, a_cos, false, b_wh, (short)0, accP, false, false);
            accQ = __builtin_amdgcn_wmma_f32_16x16x4_f32(
                false, a_sin, false, b_wh, (short)0, accQ, false, false);
        }

        // ---- flip buffers: TDM for nxt must be done, A stores visible
        if (wave == 0) {
            __builtin_amdgcn_s_wait_tensorcnt(0);
        }
        __syncthreads();
    }

    // ---- fused epilogue: tiny input GEMM (K = n_inp = 28), phase update, mod 2*pi
    // C/D layout: VGPR v, lane l -> (m = v + 8*(l>>4), n = l & 15)
    const int n_loc = lane & 15;
    const int m_hi  = (lane >> 4) << 3;
#pragma unroll
    for (int v = 0; v < 8; ++v) {
        const int b = mt * 16 + v + m_hi;
        const int i = n0 + nt * 16 + n_loc;

        const float st = state[b * n_hid + i];
        float sn, cs;
        __sincosf(st, &sn, &cs);

        // inp[b,i] = x[b,:] . Wi_w[i,:] + Wi_b[i]
        float inp = Wi_b[i];
        const float* xr = x + b * n_inp;
        const float* wr = Wi_w + (size_t)i * n_inp;
        for (int k = 0; k < n_inp; ++k) {
            inp = fmaf(xr[k], wr[k], inp);
        }

        float val = sn * accP[v] - cs * accQ[v] + inp + omega[i] + st;
        // jnp.remainder(val, 2*pi): result in [0, 2*pi)
        float r = val - floorf(val * INV_TWO_PI_F) * TWO_PI_F;
        out[b * n_hid + i] = r;
    }
}

extern "C" void kernel_launch(void* const* d_in, const int* in_sizes, int n_in,
                              void* d_out, int out_size, void* d_ws, size_t ws_size,
                              hipStream_t stream) {
    const float* x     = (const float*)d_in[0];
    const float* state = (const float*)d_in[1];
    const float* Wi_w  = (const float*)d_in[2];
    const float* Wi_b  = (const float*)d_in[3];
    const float* Wh    = (const float*)d_in[4];
    const float* omega = (const float*)d_in[5];
    float* out = (float*)d_out;

    const int n_hid = in_sizes[5];             // 2048
    const int Bsz   = in_sizes[1] / n_hid;     // 32
    const int n_inp = in_sizes[0] / Bsz;       // 28

    dim3 grid(n_hid / NBLK);                   // 32 workgroups, each owns 64 Wh rows
    dim3 block(256);                           // 8 wave32 waves
    kuramoto_wmma_kernel<<<grid, block, 0, stream>>>(
        x, state, Wi_w, Wi_b, Wh, omega, out, Bsz, n_inp, n_hid);
}